// EGONN_33225867001969
// MI455X (gfx1250) — compile-verified
//
#include <hip/hip_runtime.h>
#include <math.h>

#define NN   65536
#define EE   655360
#define GG   256
#define EGOC 64
#define SS   (GG*EGOC)
#define HH   128
#define PP   64
#define OUTC 10
#define LL   3
#define EPSF 1e-5f

typedef __attribute__((ext_vector_type(2))) float v2f;
typedef __attribute__((ext_vector_type(8))) float v8f;

// ---------- utility fills ----------
__global__ void k_fill_f(float* __restrict__ p, int n, float v) {
    int i = blockIdx.x * blockDim.x + threadIdx.x;
    if (i < n) p[i] = v;
}
__global__ void k_zero_i(int* __restrict__ p, int n) {
    int i = blockIdx.x * blockDim.x + threadIdx.x;
    if (i < n) p[i] = 0;
}

// float atomic max via int ordering trick (works for all finite floats)
__device__ inline void atomicMaxF(float* addr, float v) {
    if (!(v < 0.0f)) atomicMax((int*)addr, __float_as_int(v));
    else             atomicMin((unsigned int*)addr, __float_as_uint(v));
}

// ---------- edge scatter-add: agg[dst] += h[src] ----------
// one wave per edge; lane handles a 4-float feature chunk (b128 gather + 4 atomics)
__global__ void k_scatter_add(float* __restrict__ agg, const float* __restrict__ h,
                              const int* __restrict__ src, const int* __restrict__ dst) {
    int gid = blockIdx.x * blockDim.x + threadIdx.x;   // EE*32 = 20.97M < 2^31
    if (gid >= EE * 32) return;
    int e = gid >> 5;                 // wave-uniform edge id
    int f = (gid & 31) << 2;          // lane's 4-feature chunk
    const float4 v = *(const float4*)(h + src[e] * HH + f);
    float* a = agg + dst[e] * HH + f;
    atomicAdd(a + 0, v.x);
    atomicAdd(a + 1, v.y);
    atomicAdd(a + 2, v.z);
    atomicAdd(a + 3, v.w);
}

// ---------- fused GraphConv GEMM: out = relu(agg@wrel + hin@wroot + b) ----------
// block = 256 threads = 8 waves; block computes 64 rows x 128 cols.
// wave w -> cols [16w,16w+16); each wave register-blocks 4 row tiles so every
// B-fragment pair feeds 8 WMMAs (amortizes LDS reads and weight staging).
__global__ __launch_bounds__(256)
void k_conv_gemm(float* __restrict__ out, const float* __restrict__ agg,
                 const float* __restrict__ hin, const float* __restrict__ wrel,
                 const float* __restrict__ wroot, const float* __restrict__ bias) {
    __shared__ float sW[2][32][HH];   // K-slab of both weight matrices (32 KB)
    __shared__ float sA[2][64][33];   // A tiles, padded stride 33 (16.5 KB)
    const int tid  = threadIdx.x;
    const int lane = tid & 31;
    const int wave = tid >> 5;
    const int row0 = blockIdx.x * 64;
    const int mrow  = lane & 15;
    const int khalf = (lane & 16) ? 2 : 0;     // lanes 16-31 hold K=2,3 of each K4 group
    const int ncol  = wave * 16 + (lane & 15);
    v8f acc[4] = { {}, {}, {}, {} };
    for (int ks = 0; ks < HH; ks += 32) {
        for (int i = tid; i < 32 * HH; i += 256) {
            int kk = i >> 7, cc = i & 127;
            sW[0][kk][cc] = wrel [(ks + kk) * HH + cc];
            sW[1][kk][cc] = wroot[(ks + kk) * HH + cc];
        }
        for (int i = tid; i < 64 * 32; i += 256) {
            int r = i >> 5, kk = i & 31;
            sA[0][r][kk] = agg[(row0 + r) * HH + ks + kk];
            sA[1][r][kk] = hin[(row0 + r) * HH + ks + kk];
        }
        __syncthreads();
#pragma unroll
        for (int k4 = 0; k4 < 32; k4 += 4) {
            v2f b0 = { sW[0][k4 + khalf][ncol], sW[0][k4 + khalf + 1][ncol] };
            v2f b1 = { sW[1][k4 + khalf][ncol], sW[1][k4 + khalf + 1][ncol] };
#pragma unroll
            for (int t = 0; t < 4; ++t) {
                int rr = t * 16 + mrow;
                v2f a0 = { sA[0][rr][k4 + khalf], sA[0][rr][k4 + khalf + 1] };
                acc[t] = __builtin_amdgcn_wmma_f32_16x16x4_f32(false, a0, false, b0, (short)0, acc[t], false, false);
                v2f a1 = { sA[1][rr][k4 + khalf], sA[1][rr][k4 + khalf + 1] };
                acc[t] = __builtin_amdgcn_wmma_f32_16x16x4_f32(false, a1, false, b1, (short)0, acc[t], false, false);
            }
        }
        __syncthreads();
    }
    float bv = bias[ncol];
#pragma unroll
    for (int t = 0; t < 4; ++t) {
#pragma unroll
        for (int r = 0; r < 8; ++r) {
            int M = t * 16 + r + ((lane & 16) ? 8 : 0);   // C layout: VGPR r -> M=r / r+8
            out[(row0 + M) * HH + ncol] = fmaxf(acc[t][r] + bv, 0.f);
        }
    }
}

// ---------- down projection: xi = h @ down_w  ([N,128]x[128,64]) ----------
// block = 256 threads = 8 waves; 32 rows x 64 cols per block
// waves 0-3 -> rows 0-15 (col strips), waves 4-7 -> rows 16-31
__global__ __launch_bounds__(256)
void k_down(float* __restrict__ xi, const float* __restrict__ hin,
            const float* __restrict__ dw) {
    __shared__ float sW[HH][PP];      // 32 KB
    __shared__ float sA[32][129];     // padded (16.5 KB)
    const int tid  = threadIdx.x;
    const int lane = tid & 31;
    const int wave = tid >> 5;
    const int row0 = blockIdx.x * 32;
    const int rowt = (wave >> 2) * 16;
    const int mrow  = rowt + (lane & 15);
    const int khalf = (lane & 16) ? 2 : 0;
    const int ncol  = (wave & 3) * 16 + (lane & 15);
    for (int i = tid; i < HH * PP; i += 256) sW[i >> 6][i & 63] = dw[i];
    for (int i = tid; i < 32 * HH; i += 256) sA[i >> 7][i & 127] = hin[(row0 + (i >> 7)) * HH + (i & 127)];
    __syncthreads();
    v8f c = {};
#pragma unroll
    for (int k4 = 0; k4 < HH; k4 += 4) {
        v2f a = { sA[mrow][k4 + khalf], sA[mrow][k4 + khalf + 1] };
        v2f b = { sW[k4 + khalf][ncol], sW[k4 + khalf + 1][ncol] };
        c = __builtin_amdgcn_wmma_f32_16x16x4_f32(false, a, false, b, (short)0, c, false, false);
    }
#pragma unroll
    for (int r = 0; r < 8; ++r) {
        int M = rowt + r + ((lane & 16) ? 8 : 0);
        xi[(row0 + M) * PP + ncol] = c[r];
    }
}

// ---------- ego pooling: pooled[ego_g] += xi ; seg_batch[ego_g] = max(batch) ----------
__global__ void k_pool(float* __restrict__ pooled, int* __restrict__ segb,
                       const float* __restrict__ xi, const int* __restrict__ batch,
                       const int* __restrict__ ego) {
    int gid = blockIdx.x * blockDim.x + threadIdx.x;
    if (gid >= NN * PP) return;
    int i = gid >> 6, p = gid & 63;
    int b = batch[i];
    int s = ego[i] + b * EGOC;
    atomicAdd(&pooled[s * PP + p], xi[gid]);
    if (p == 0) atomicMax(&segb[s], b);
}

// ---------- softmax aggregation over ego segments within each graph ----------
__global__ void k_seg_max(float* __restrict__ m, const float* __restrict__ pooled,
                          const int* __restrict__ segb) {
    int gid = blockIdx.x * blockDim.x + threadIdx.x;
    if (gid >= SS * PP) return;
    int s = gid >> 6, p = gid & 63;
    atomicMaxF(&m[segb[s] * PP + p], pooled[gid]);
}
__global__ void k_seg_denom(float* __restrict__ denom, const float* __restrict__ pooled,
                            const float* __restrict__ m, const int* __restrict__ segb) {
    int gid = blockIdx.x * blockDim.x + threadIdx.x;
    if (gid >= SS * PP) return;
    int s = gid >> 6, p = gid & 63;
    int g = segb[s];
    atomicAdd(&denom[g * PP + p], expf(pooled[gid] - m[g * PP + p]));
}
__global__ void k_seg_out(float* __restrict__ gv, const float* __restrict__ pooled,
                          const float* __restrict__ m, const float* __restrict__ denom,
                          const int* __restrict__ segb) {
    int gid = blockIdx.x * blockDim.x + threadIdx.x;
    if (gid >= SS * PP) return;
    int s = gid >> 6, p = gid & 63;
    int g = segb[s];
    float pv = pooled[gid];
    float e  = expf(pv - m[g * PP + p]);
    atomicAdd(&gv[g * PP + p], (e / denom[g * PP + p]) * pv);
}

// ---------- MLP head: Linear(64->20) -> BatchNorm(G rows) -> ReLU -> Linear(20->10) ----------
__global__ __launch_bounds__(256)
void k_mlp(float* __restrict__ out, const float* __restrict__ gv,
           const float* __restrict__ w1, const float* __restrict__ b1,
           const float* __restrict__ gamma, const float* __restrict__ beta,
           const float* __restrict__ w2, const float* __restrict__ b2) {
    __shared__ float sz[GG][2 * OUTC];
    __shared__ float smu[2 * OUTC], srstd[2 * OUTC];
    const int r = threadIdx.x;        // one row (graph) per thread, 256 threads
    float zr[2 * OUTC];
#pragma unroll
    for (int j = 0; j < 2 * OUTC; ++j) zr[j] = b1[j];
    for (int k = 0; k < PP; ++k) {
        float gvk = gv[r * PP + k];
#pragma unroll
        for (int j = 0; j < 2 * OUTC; ++j) zr[j] += gvk * w1[k * 2 * OUTC + j];
    }
#pragma unroll
    for (int j = 0; j < 2 * OUTC; ++j) sz[r][j] = zr[j];
    __syncthreads();
    if (r < 2 * OUTC) {
        float s = 0.f, s2 = 0.f;
        for (int i = 0; i < GG; ++i) { float v = sz[i][r]; s += v; s2 += v * v; }
        float mu  = s / (float)GG;
        float var = s2 / (float)GG - mu * mu;
        smu[r]   = mu;
        srstd[r] = rsqrtf(var + EPSF);
    }
    __syncthreads();
    float a[2 * OUTC];
#pragma unroll
    for (int j = 0; j < 2 * OUTC; ++j) {
        float v = (sz[r][j] - smu[j]) * srstd[j] * gamma[j] + beta[j];
        a[j] = fmaxf(v, 0.f);
    }
#pragma unroll
    for (int o = 0; o < OUTC; ++o) {
        float acc = b2[o];
#pragma unroll
        for (int j = 0; j < 2 * OUTC; ++j) acc += a[j] * w2[j * OUTC + o];
        out[r * OUTC + o] = acc;
    }
}

extern "C" void kernel_launch(void* const* d_in, const int* in_sizes, int n_in,
                              void* d_out, int out_size, void* d_ws, size_t ws_size,
                              hipStream_t stream) {
    const float* x      = (const float*)d_in[0];
    const int*   edge   = (const int*)  d_in[1];
    const int*   batch  = (const int*)  d_in[2];
    const int*   ego    = (const int*)  d_in[3];
    const float* w_root = (const float*)d_in[4];
    const float* w_rel  = (const float*)d_in[5];
    const float* b_conv = (const float*)d_in[6];
    const float* down_w = (const float*)d_in[7];
    const float* w1     = (const float*)d_in[8];
    const float* b1     = (const float*)d_in[9];
    const float* gamma  = (const float*)d_in[10];
    const float* beta   = (const float*)d_in[11];
    const float* w2     = (const float*)d_in[12];
    const float* b2     = (const float*)d_in[13];
    const int* src = edge;
    const int* dst = edge + EE;
    float* outp = (float*)d_out;

    // workspace carve-up (floats)
    float* ws = (float*)d_ws;
    size_t off = 0;
    float* h_a    = ws + off; off += (size_t)NN * HH;
    float* h_b    = ws + off; off += (size_t)NN * HH;
    float* agg    = ws + off; off += (size_t)NN * HH;
    float* xi     = ws + off; off += (size_t)NN * PP;
    float* pooled = ws + off; off += (size_t)SS * PP;
    float* mbuf   = ws + off; off += (size_t)GG * PP;
    float* denom  = ws + off; off += (size_t)GG * PP;
    float* gv     = ws + off; off += (size_t)GG * PP;
    int*   segb   = (int*)(ws + off);

    // ---- GraphConv layers ----
    const float* hin = x;
    float* bufs[3] = { h_a, h_b, h_a };
    for (int l = 0; l < LL; ++l) {
        k_fill_f<<<(NN * HH + 255) / 256, 256, 0, stream>>>(agg, NN * HH, 0.f);
        k_scatter_add<<<(EE * 32 + 255) / 256, 256, 0, stream>>>(agg, hin, src, dst);
        k_conv_gemm<<<NN / 64, 256, 0, stream>>>(bufs[l], agg, hin,
                                                 w_rel  + (size_t)l * HH * HH,
                                                 w_root + (size_t)l * HH * HH,
                                                 b_conv + (size_t)l * HH);
        hin = bufs[l];
    }

    // ---- down projection ----
    k_down<<<NN / 32, 256, 0, stream>>>(xi, hin, down_w);

    // ---- ego pooling + per-graph softmax aggregation ----
    k_fill_f<<<(SS * PP + 255) / 256, 256, 0, stream>>>(pooled, SS * PP, 0.f);
    k_zero_i<<<(SS + 255) / 256, 256, 0, stream>>>(segb, SS);
    k_fill_f<<<(GG * PP + 255) / 256, 256, 0, stream>>>(mbuf, GG * PP, -INFINITY);
    k_fill_f<<<(GG * PP + 255) / 256, 256, 0, stream>>>(denom, GG * PP, 0.f);
    k_fill_f<<<(GG * PP + 255) / 256, 256, 0, stream>>>(gv, GG * PP, 0.f);

    k_pool<<<(NN * PP + 255) / 256, 256, 0, stream>>>(pooled, segb, xi, batch, ego);
    k_seg_max  <<<(SS * PP + 255) / 256, 256, 0, stream>>>(mbuf, pooled, segb);
    k_seg_denom<<<(SS * PP + 255) / 256, 256, 0, stream>>>(denom, pooled, mbuf, segb);
    k_seg_out  <<<(SS * PP + 255) / 256, 256, 0, stream>>>(gv, pooled, mbuf, denom, segb);

    // ---- MLP head ----
    k_mlp<<<1, GG, 0, stream>>>(outp, gv, w1, b1, gamma, beta, w2, b2);
}